// GNN_34402688041506
// MI455X (gfx1250) — compile-verified
//
#include <hip/hip_runtime.h>

#define N_NODES 50000
#define N_EDGES 250000
#define DIM     300
#define NLAYER  5
#define NGRAPH  1000
#define NCLASS  6
#define EDGE_F  7
#define BN_EPS  1e-5f

typedef __attribute__((ext_vector_type(2))) float v2f;
typedef __attribute__((ext_vector_type(8))) float v8f;

// ---------------------------------------------------------------- utilities
__global__ void k_zero(float* __restrict__ p, int n) {
    int i = blockIdx.x * blockDim.x + threadIdx.x;
    if (i < n) p[i] = 0.0f;
}

// h[n,d] = node_emb[x[n], d]
__global__ void k_init_h(const float* __restrict__ emb, const int* __restrict__ x,
                         float* __restrict__ h, int total, int D) {
    int i = blockIdx.x * blockDim.x + threadIdx.x;
    if (i >= total) return;
    int n = i / D;
    int d = i - n * D;
    h[i] = emb[(size_t)x[n] * D + d];
}

// agg[i] = (1 + eps[l]) * h[i]   (scatter target pre-seeded with the self term)
__global__ void k_combine(const float* __restrict__ h, const float* __restrict__ eps,
                          int l, float* __restrict__ agg, int total) {
    int i = blockIdx.x * blockDim.x + threadIdx.x;
    if (i >= total) return;
    float f = 1.0f + eps[l];
    agg[i] = f * h[i];
}

// Pair-interleave weights: Wp[(k/2)*Nc*2 + n*2 + (k&1)] = W[k*Nc + n]
// -> a lane's two K-values for one B column become one contiguous float2.
__global__ void k_pack_w(const float* __restrict__ W, float* __restrict__ Wp, int K, int Nc) {
    int i = blockIdx.x * blockDim.x + threadIdx.x;
    if (i >= K * Nc) return;
    int k = i / Nc;
    int n = i - k * Nc;
    Wp[(size_t)(k >> 1) * Nc * 2 + n * 2 + (k & 1)] = W[i];
}

// ------------------------------------------------------- edge encode+scatter
// One wave per edge: msg = relu(h[src] + edge_attr@W + b); atomicAdd into agg[dst].
__global__ void k_edge(const float* __restrict__ h, const int* __restrict__ src,
                       const int* __restrict__ dst, const float* __restrict__ edge_attr,
                       const float* __restrict__ eW, const float* __restrict__ eb,
                       float* __restrict__ agg, int E) {
    __shared__ float sW[EDGE_F * DIM];
    __shared__ float sb[DIM];
    for (int i = threadIdx.x; i < EDGE_F * DIM; i += blockDim.x) sW[i] = eW[i];
    for (int i = threadIdx.x; i < DIM; i += blockDim.x) sb[i] = eb[i];
    __syncthreads();

    int wave = threadIdx.x >> 5;
    int lane = threadIdx.x & 31;
    int e = blockIdx.x * (blockDim.x >> 5) + wave;
    if (e >= E) return;

    int s = src[e];
    int t = dst[e];
    float a[EDGE_F];
#pragma unroll
    for (int f = 0; f < EDGE_F; ++f) a[f] = edge_attr[(size_t)e * EDGE_F + f];

    const float* hs = h + (size_t)s * DIM;
    float* ag = agg + (size_t)t * DIM;
    for (int d = lane; d < DIM; d += 32) {
        float m = sb[d];
#pragma unroll
        for (int f = 0; f < EDGE_F; ++f) m += a[f] * sW[f * DIM + d];
        m += hs[d];
        m = fmaxf(m, 0.0f);
        atomicAdd(&ag[d], m);
    }
}

// ------------------------------------------------------------- WMMA GEMM
// C[M,Nc] = Aop(A[M,K]) @ B[K,Nc] + bias, optional post-ReLU.
// B is pre-packed pair-interleaved (see k_pack_w): Bp row kp holds K-rows {2kp, 2kp+1}
// as float2 per column -> one global_load_b64 per B operand per lane.
// AOP: 0 identity, 1 relu(a*scale[k]+shift[k]) (fused BN+ReLU), 2 relu(a).
// One wave -> 32x32 macro-tile = 4 accumulators, 4 v_wmma_f32_16x16x4_f32 per K-step.
// Partial tiles: load addresses CLAMPED (row r of A only feeds row r of D, col n of
// B only feeds col n of D), stores guarded -> no selects in the loop.
template <int AOP, int POST>
__global__ void k_gemm_wmma(const float* __restrict__ A, const v2f* __restrict__ Bp,
                            const float* __restrict__ bias, float* __restrict__ C,
                            int M, int Nc, int K,
                            const float* __restrict__ scale, const float* __restrict__ shift) {
    __shared__ float sScale[DIM + 4];
    __shared__ float sShift[DIM + 4];
    if (AOP == 1) {
        for (int i = threadIdx.x; i < K; i += blockDim.x) {
            sScale[i] = scale[i];
            sShift[i] = shift[i];
        }
        __syncthreads();
    }

    int wave = threadIdx.x >> 5;
    int lane = threadIdx.x & 31;
    int tM = blockIdx.x * (blockDim.x >> 5) + wave;   // 32-row macro-tile index
    int tN = blockIdx.y;                              // 32-col macro-tile index
    if (tM * 32 >= M) return;                         // wave-uniform exit

    int l15   = lane & 15;
    int khalf = (lane >> 4) << 1;                     // 0 or 2
    int row0 = tM * 32 + l15;
    int row1 = row0 + 16;
    int c0   = tN * 32 + l15;
    int c1   = c0 + 16;
    int r0c = row0 < M ? row0 : M - 1;                // clamped addresses
    int r1c = row1 < M ? row1 : M - 1;
    int c0c = c0 < Nc ? c0 : Nc - 1;
    int c1c = c1 < Nc ? c1 : Nc - 1;

    // A rows as float2 (8B aligned: K even, khalf even)
    const v2f* A0 = (const v2f*)(A + (size_t)r0c * K + khalf);
    const v2f* A1 = (const v2f*)(A + (size_t)r1c * K + khalf);
    // Packed B: base offset selects this half-wave's K-pair row
    const v2f* B0 = Bp + (size_t)(khalf >> 1) * Nc + c0c;
    const v2f* B1 = Bp + (size_t)(khalf >> 1) * Nc + c1c;

    v8f acc00 = {0.f, 0.f, 0.f, 0.f, 0.f, 0.f, 0.f, 0.f};
    v8f acc01 = acc00, acc10 = acc00, acc11 = acc00;

    for (int k0 = 0; k0 < K; k0 += 4) {
        int kh = k0 >> 1;                 // float2 index along K
        int ka = k0 + khalf;              // scalar K index (for BN params)

        v2f av0 = A0[kh];
        v2f av1 = A1[kh];
        if (AOP == 1) {
            float sx = sScale[ka], sy = sScale[ka + 1];
            float fx = sShift[ka], fy = sShift[ka + 1];
            av0.x = fmaxf(av0.x * sx + fx, 0.0f);
            av0.y = fmaxf(av0.y * sy + fy, 0.0f);
            av1.x = fmaxf(av1.x * sx + fx, 0.0f);
            av1.y = fmaxf(av1.y * sy + fy, 0.0f);
        } else if (AOP == 2) {
            av0.x = fmaxf(av0.x, 0.0f);
            av0.y = fmaxf(av0.y, 0.0f);
            av1.x = fmaxf(av1.x, 0.0f);
            av1.y = fmaxf(av1.y, 0.0f);
        }

        v2f bv0 = B0[(size_t)kh * Nc];    // global_load_b64, 128B coalesced/16 lanes
        v2f bv1 = B1[(size_t)kh * Nc];

        acc00 = __builtin_amdgcn_wmma_f32_16x16x4_f32(false, av0, false, bv0, (short)0, acc00, false, false);
        acc01 = __builtin_amdgcn_wmma_f32_16x16x4_f32(false, av0, false, bv1, (short)0, acc01, false, false);
        acc10 = __builtin_amdgcn_wmma_f32_16x16x4_f32(false, av1, false, bv0, (short)0, acc10, false, false);
        acc11 = __builtin_amdgcn_wmma_f32_16x16x4_f32(false, av1, false, bv1, (short)0, acc11, false, false);
    }

    // C/D layout: VGPR r -> row r (lanes 0-15) / row r+8 (lanes 16-31), col = lane&15
    int rSub = (lane >> 4) << 3;   // 0 or 8
    {
        const int rowOffs[4] = {0, 0, 16, 16};
        const int colIdxs[4] = {c0, c1, c0, c1};
        v8f accs[4] = {acc00, acc01, acc10, acc11};
#pragma unroll
        for (int tIdx = 0; tIdx < 4; ++tIdx) {
            int col = colIdxs[tIdx];
            if (col < Nc) {
                float bb = bias[col];
#pragma unroll
                for (int r = 0; r < 8; ++r) {
                    int orow = tM * 32 + rowOffs[tIdx] + rSub + r;
                    if (orow < M) {
                        float v = accs[tIdx][r] + bb;
                        if (POST) v = fmaxf(v, 0.0f);
                        C[(size_t)orow * Nc + col] = v;
                    }
                }
            }
        }
    }
}

// ------------------------------------------------------------- batch norm
__global__ void k_bn_stats(const float* __restrict__ z, float* __restrict__ sum,
                           float* __restrict__ sumsq, int Nrows, int D, int rowsPerBlock) {
    int c = threadIdx.x;
    if (c >= D) return;
    int r0 = blockIdx.x * rowsPerBlock;
    int r1 = r0 + rowsPerBlock;
    if (r1 > Nrows) r1 = Nrows;
    float s = 0.0f, q = 0.0f;
    for (int r = r0; r < r1; ++r) {
        float v = z[(size_t)r * D + c];
        s += v;
        q += v * v;
    }
    atomicAdd(&sum[c], s);
    atomicAdd(&sumsq[c], q);
}

__global__ void k_bn_finalize(const float* __restrict__ sum, const float* __restrict__ sumsq,
                              const float* __restrict__ g, const float* __restrict__ b,
                              float* __restrict__ scale, float* __restrict__ shift,
                              int Nrows, int D) {
    int c = blockIdx.x * blockDim.x + threadIdx.x;
    if (c >= D) return;
    float inv = 1.0f / (float)Nrows;
    float m = sum[c] * inv;
    float var = sumsq[c] * inv - m * m;       // biased variance (matches jnp.var)
    float sc = g[c] * rsqrtf(var + BN_EPS);
    scale[c] = sc;
    shift[c] = b[c] - m * sc;
}

__global__ void k_bn_apply(const float* __restrict__ z, const float* __restrict__ scale,
                           const float* __restrict__ shift, float* __restrict__ out,
                           int total, int D, int doRelu) {
    int i = blockIdx.x * blockDim.x + threadIdx.x;
    if (i >= total) return;
    int c = i % D;
    float v = z[i] * scale[c] + shift[c];
    if (doRelu) v = fmaxf(v, 0.0f);
    out[i] = v;
}

// ------------------------------------------------------------- pooling / head
__global__ void k_pool(const float* __restrict__ h, const int* __restrict__ batch,
                       float* __restrict__ hgsum, float* __restrict__ cnt, int N, int D) {
    int wave = threadIdx.x >> 5;
    int lane = threadIdx.x & 31;
    int n = blockIdx.x * (blockDim.x >> 5) + wave;
    if (n >= N) return;
    int g = batch[n];
    if (lane == 0) atomicAdd(&cnt[g], 1.0f);
    const float* hn = h + (size_t)n * D;
    float* hs = hgsum + (size_t)g * D;
    for (int d = lane; d < D; d += 32) atomicAdd(&hs[d], hn[d]);
}

__global__ void k_pool_div(float* __restrict__ hg, const float* __restrict__ cnt,
                           int G, int D) {
    int i = blockIdx.x * blockDim.x + threadIdx.x;
    if (i >= G * D) return;
    hg[i] = hg[i] / fmaxf(cnt[i / D], 1.0f);
}

__global__ void k_pred(const float* __restrict__ graph, const float* __restrict__ W,
                       const float* __restrict__ b, float* __restrict__ out,
                       int G, int D, int C) {
    int i = blockIdx.x * blockDim.x + threadIdx.x;
    if (i >= G * C) return;
    int g = i / C;
    int c = i - g * C;
    float acc = b[c];
    const float* gr = graph + (size_t)g * D;
    for (int d = 0; d < D; ++d) acc += gr[d] * W[(size_t)d * C + c];
    out[i] = acc;
}

// ---------------------------------------------------------------- driver
extern "C" void kernel_launch(void* const* d_in, const int* in_sizes, int n_in,
                              void* d_out, int out_size, void* d_ws, size_t ws_size,
                              hipStream_t stream) {
    (void)in_sizes; (void)n_in; (void)out_size; (void)ws_size;

    const int*   x         = (const int*)d_in[0];
    const int*   edge_idx  = (const int*)d_in[1];
    const int*   src       = edge_idx;             // edge_index[0,:]
    const int*   dst       = edge_idx + N_EDGES;   // edge_index[1,:]
    const float* edge_attr = (const float*)d_in[2];
    const int*   batch     = (const int*)d_in[3];
    const float* node_emb  = (const float*)d_in[4];
    const float* edge_W    = (const float*)d_in[5];
    const float* edge_b    = (const float*)d_in[6];
    const float* eps       = (const float*)d_in[7];
    const float* mlp_W1    = (const float*)d_in[8];
    const float* mlp_b1    = (const float*)d_in[9];
    const float* mlp_bn_g  = (const float*)d_in[10];
    const float* mlp_bn_b  = (const float*)d_in[11];
    const float* mlp_W2    = (const float*)d_in[12];
    const float* mlp_b2    = (const float*)d_in[13];
    const float* bn_g      = (const float*)d_in[14];
    const float* bn_b      = (const float*)d_in[15];
    const float* lin1_W    = (const float*)d_in[16];
    const float* lin1_b    = (const float*)d_in[17];
    const float* pred_W    = (const float*)d_in[18];
    const float* pred_b    = (const float*)d_in[19];

    float* out_pred  = (float*)d_out;                      // [G,C]
    float* out_graph = out_pred + NGRAPH * NCLASS;         // [G,D]
    float* out_hnode = out_graph + (size_t)NGRAPH * DIM;   // [N,D]

    const size_t ND = (size_t)N_NODES * DIM;
    const size_t WSZ = (size_t)DIM * DIM;      // 90000 (even -> keeps 8B alignment)
    float* ws    = (float*)d_ws;
    float* h     = ws;                 // [N,D]
    float* bufA  = h + ND;             // agg, later z2
    float* bufB  = bufA + ND;          // z1
    float* hg    = bufB + ND;          // [G,D]
    float* cnt   = hg + (size_t)NGRAPH * DIM;  // [G]
    float* sum   = cnt + NGRAPH;       // [D]
    float* sumsq = sum + DIM;          // [D]
    float* scl   = sumsq + DIM;        // [D]
    float* shf   = scl + DIM;          // [D]
    float* packW = shf + DIM;          // 11 * 90000 pair-interleaved weights

    const int totalND = (int)ND;
    const int nb256   = (totalND + 255) / 256;
    const int nbW     = ((int)WSZ + 255) / 256;

    // Pre-pack all GEMM weights (pair-interleaved along K)
    for (int l = 0; l < NLAYER; ++l) {
        k_pack_w<<<nbW, 256, 0, stream>>>(mlp_W1 + (size_t)l * WSZ, packW + (size_t)l * WSZ, DIM, DIM);
        k_pack_w<<<nbW, 256, 0, stream>>>(mlp_W2 + (size_t)l * WSZ, packW + (size_t)(NLAYER + l) * WSZ, DIM, DIM);
    }
    k_pack_w<<<nbW, 256, 0, stream>>>(lin1_W, packW + (size_t)(2 * NLAYER) * WSZ, DIM, DIM);

    // h = node_emb[x]
    k_init_h<<<nb256, 256, 0, stream>>>(node_emb, x, h, totalND, DIM);

    // M=50000 -> 1563 32-row macro-tiles; Nc=300 -> 10 32-col macro-tiles
    dim3 gemmGridBig((((N_NODES + 31) / 32) + 3) / 4, (DIM + 31) / 32);

    for (int l = 0; l < NLAYER; ++l) {
        // agg = (1+eps)*h, then scatter-add edge messages
        k_combine<<<nb256, 256, 0, stream>>>(h, eps, l, bufA, totalND);
        k_edge<<<(N_EDGES + 7) / 8, 256, 0, stream>>>(
            h, src, dst, edge_attr,
            edge_W + (size_t)l * EDGE_F * DIM, edge_b + (size_t)l * DIM, bufA, N_EDGES);

        // z1 = agg @ W1 + b1
        k_gemm_wmma<0, 0><<<gemmGridBig, 128, 0, stream>>>(
            bufA, (const v2f*)(packW + (size_t)l * WSZ), mlp_b1 + (size_t)l * DIM, bufB,
            N_NODES, DIM, DIM, scl, shf);

        // inner BN stats -> scale/shift
        k_zero<<<(2 * DIM + 255) / 256, 256, 0, stream>>>(sum, 2 * DIM);
        k_bn_stats<<<(N_NODES + 127) / 128, 320, 0, stream>>>(bufB, sum, sumsq, N_NODES, DIM, 128);
        k_bn_finalize<<<(DIM + 255) / 256, 256, 0, stream>>>(
            sum, sumsq, mlp_bn_g + (size_t)l * DIM, mlp_bn_b + (size_t)l * DIM,
            scl, shf, N_NODES, DIM);

        // z2 = relu(bn(z1)) @ W2 + b2   (BN+ReLU fused into A-load via LDS params)
        k_gemm_wmma<1, 0><<<gemmGridBig, 128, 0, stream>>>(
            bufB, (const v2f*)(packW + (size_t)(NLAYER + l) * WSZ), mlp_b2 + (size_t)l * DIM, bufA,
            N_NODES, DIM, DIM, scl, shf);

        // outer BN stats + apply (ReLU except last layer)
        k_zero<<<(2 * DIM + 255) / 256, 256, 0, stream>>>(sum, 2 * DIM);
        k_bn_stats<<<(N_NODES + 127) / 128, 320, 0, stream>>>(bufA, sum, sumsq, N_NODES, DIM, 128);
        k_bn_finalize<<<(DIM + 255) / 256, 256, 0, stream>>>(
            sum, sumsq, bn_g + (size_t)l * DIM, bn_b + (size_t)l * DIM,
            scl, shf, N_NODES, DIM);
        k_bn_apply<<<nb256, 256, 0, stream>>>(bufA, scl, shf, h, totalND, DIM,
                                              (l < NLAYER - 1) ? 1 : 0);
    }

    // h_node output
    hipMemcpyAsync(out_hnode, h, ND * sizeof(float), hipMemcpyDeviceToDevice, stream);

    // global mean pool
    k_zero<<<((NGRAPH * DIM + NGRAPH) + 255) / 256, 256, 0, stream>>>(hg, NGRAPH * DIM + NGRAPH);
    k_pool<<<(N_NODES + 7) / 8, 256, 0, stream>>>(h, batch, hg, cnt, N_NODES, DIM);
    k_pool_div<<<(NGRAPH * DIM + 255) / 256, 256, 0, stream>>>(hg, cnt, NGRAPH, DIM);

    // graph = relu(relu(hg) @ lin1_W + lin1_b)   (AOP=2, post-ReLU)
    dim3 gemmGridSmall((((NGRAPH + 31) / 32) + 3) / 4, (DIM + 31) / 32);
    k_gemm_wmma<2, 1><<<gemmGridSmall, 128, 0, stream>>>(
        hg, (const v2f*)(packW + (size_t)(2 * NLAYER) * WSZ), lin1_b, out_graph,
        NGRAPH, DIM, DIM, scl, shf);

    // out = graph @ pred_W + pred_b  (tiny: VALU)
    k_pred<<<(NGRAPH * NCLASS + 255) / 256, 256, 0, stream>>>(
        out_graph, pred_W, pred_b, out_pred, NGRAPH, DIM, NCLASS);
}